// UnaryStack_42477226557548
// MI455X (gfx1250) — compile-verified
//
#include <hip/hip_runtime.h>
#include <hip/hip_bf16.h>

typedef __attribute__((ext_vector_type(2))) float v2f;
typedef __attribute__((ext_vector_type(4))) float v4f;
typedef __attribute__((ext_vector_type(8))) float v8f;

#define D_MODEL 512
#define STACK_S 64
#define BATCH   1024
#define PLANE   (BATCH * D_MODEL)   // 524288 floats per coefficient plane

__device__ __forceinline__ float sigf(float x) {
    return 1.0f / (1.0f + __expf(-x));
}

// ---------------------------------------------------------------------------
// Kernel 1: six fused GEMMs (fp32 WMMA 16x16x4) + sigmoid + 3-way softmax,
// emitting 4 pre-combined coefficient planes:
//   plane 0: cPush = gate * Pg      (multiplies stack[s-1])
//   plane 1: cPop  = gate * Qg      (multiplies stack[s+1])
//   plane 2: cNoop = gate * Ng      (multiplies stack[s])
//   plane 3: cPI   = Pg * data * input   (push term of row 0)
// One wave per (16-row batch tile, 16-col e slice); 2048 waves total.
// ---------------------------------------------------------------------------
__global__ __launch_bounds__(256) void gates_wmma_kernel(
    const float* __restrict__ inp,
    const float* __restrict__ Wg, const float* __restrict__ bg,
    const float* __restrict__ Wi, const float* __restrict__ bi,
    const float* __restrict__ Wd, const float* __restrict__ bd,
    const float* __restrict__ Wp, const float* __restrict__ bp,
    const float* __restrict__ Wq, const float* __restrict__ bq,
    const float* __restrict__ Wn, const float* __restrict__ bn,
    float* __restrict__ coef)
{
    const int lane = threadIdx.x & 31;
    const int wave = threadIdx.x >> 5;
    const int tile = blockIdx.x * 8 + wave;     // 0..2047
    const int mt   = tile >> 5;                 // 0..63  batch tile
    const int et   = tile & 31;                 // 0..31  e slice
    const int half = lane >> 4;                 // 0 or 1
    const int l15  = lane & 15;
    const int koff = half << 1;                 // 0 or 2 (K sub-offset per ISA layout)

    const int row = mt * 16 + l15;              // A-matrix row (batch index)
    const int e   = et * 16 + l15;              // B-matrix column (output feature)

    const float* Aptr = inp + row * D_MODEL + koff;
    const float* Wm[6]  = {Wg, Wi, Wd, Wp, Wq, Wn};
    const float* bv6[6] = {bg, bi, bd, bp, bq, bn};
    const float* Bptr[6];
#pragma unroll
    for (int j = 0; j < 6; ++j) Bptr[j] = Wm[j] + e * D_MODEL + koff;

    v8f acc[6];
#pragma unroll
    for (int j = 0; j < 6; ++j) acc[j] = (v8f){0.f,0.f,0.f,0.f,0.f,0.f,0.f,0.f};

    // K loop: 128 steps of K=4; A frag reused across all 6 weight matrices.
    for (int k = 0; k < D_MODEL; k += 4) {
        v2f a = *(const v2f*)(Aptr + k);
#pragma unroll
        for (int j = 0; j < 6; ++j) {
            v2f b = *(const v2f*)(Bptr[j] + k);
            acc[j] = __builtin_amdgcn_wmma_f32_16x16x4_f32(
                false, a, false, b, (short)0, acc[j], false, false);
        }
    }

    float bias[6];
#pragma unroll
    for (int j = 0; j < 6; ++j) bias[j] = bv6[j][e];

    // D layout: VGPR j holds element (m = mt*16 + j + 8*half, n = e).
#pragma unroll
    for (int j = 0; j < 8; ++j) {
        const int m = mt * 16 + j + half * 8;
        const float g  = sigf(acc[0][j] + bias[0]);   // gate
        const float ig = sigf(acc[1][j] + bias[1]);   // input gate
        const float dg = sigf(acc[2][j] + bias[2]);   // data gate
        const float sp = sigf(acc[3][j] + bias[3]);   // push logit (post-sigmoid)
        const float sq = sigf(acc[4][j] + bias[4]);   // pop
        const float sn = sigf(acc[5][j] + bias[5]);   // noop
        // softmax over the three sigmoided gates
        const float mx  = fmaxf(sp, fmaxf(sq, sn));
        const float ep  = __expf(sp - mx);
        const float eq  = __expf(sq - mx);
        const float en  = __expf(sn - mx);
        const float inv = 1.0f / (ep + eq + en);
        const float Pg = ep * inv, Qg = eq * inv, Ng = en * inv;

        const int idx = m * D_MODEL + e;
        coef[0 * PLANE + idx] = g * Pg;          // cPush
        coef[1 * PLANE + idx] = g * Qg;          // cPop
        coef[2 * PLANE + idx] = g * Ng;          // cNoop
        coef[3 * PLANE + idx] = Pg * dg * ig;    // cPI
    }
}

// ---------------------------------------------------------------------------
// Kernel 2: memory-bound shift-and-blend over the stack.
// One thread owns (b, e..e+3); float4 rolling window over s; each stack
// element is loaded exactly once, fully coalesced (consecutive lanes ->
// consecutive e).
// ---------------------------------------------------------------------------
__global__ __launch_bounds__(256) void stack_stream_kernel(
    const float* __restrict__ stack,
    const float* __restrict__ coef,
    float* __restrict__ out)
{
    const int t = blockIdx.x * 256 + threadIdx.x;   // 0..131071
    const int b = t >> 7;                           // batch row
    const int e = (t & 127) << 2;                   // feature (x4)
    const int idx = b * D_MODEL + e;

    const v4f cPush = *(const v4f*)(coef + 0 * PLANE + idx);
    const v4f cPop  = *(const v4f*)(coef + 1 * PLANE + idx);
    const v4f cNoop = *(const v4f*)(coef + 2 * PLANE + idx);
    const v4f cPI   = *(const v4f*)(coef + 3 * PLANE + idx);

    const v4f* sp = (const v4f*)(stack + (size_t)b * STACK_S * D_MODEL + e);
    v4f*       op = (v4f*)(out  + (size_t)b * STACK_S * D_MODEL + e);
    const int stride = D_MODEL / 4;                 // 128 v4f per s step

    v4f vm = sp[0];
    v4f vc = sp[stride];
    // s = 0: push term replaced by cPI
    op[0] = cPI + cPop * vc + cNoop * vm;

#pragma unroll 8
    for (int s = 1; s < STACK_S - 1; ++s) {
        v4f vp = sp[(s + 1) * stride];
        op[s * stride] = cPush * vm + cPop * vp + cNoop * vc;
        vm = vc;
        vc = vp;
    }
    // s = 63: pop term is zero (st[64] == 0)
    op[(STACK_S - 1) * stride] = cPush * vm + cNoop * vc;
}

extern "C" void kernel_launch(void* const* d_in, const int* in_sizes, int n_in,
                              void* d_out, int out_size, void* d_ws, size_t ws_size,
                              hipStream_t stream) {
    const float* inp   = (const float*)d_in[0];
    const float* stack = (const float*)d_in[1];
    const float* Wg = (const float*)d_in[2];  const float* bg = (const float*)d_in[3];
    const float* Wi = (const float*)d_in[4];  const float* bi = (const float*)d_in[5];
    const float* Wd = (const float*)d_in[6];  const float* bd = (const float*)d_in[7];
    const float* Wp = (const float*)d_in[8];  const float* bp = (const float*)d_in[9];
    const float* Wq = (const float*)d_in[10]; const float* bq = (const float*)d_in[11];
    const float* Wn = (const float*)d_in[12]; const float* bn = (const float*)d_in[13];
    float* out  = (float*)d_out;
    float* coef = (float*)d_ws;   // 4 planes * 1024 * 512 floats = 8 MB

    // Kernel 1: 2048 waves (64 batch tiles x 32 e-slices), 8 waves/block.
    gates_wmma_kernel<<<256, 256, 0, stream>>>(
        inp, Wg, bg, Wi, bi, Wd, bd, Wp, bp, Wq, bq, Wn, bn, coef);

    // Kernel 2: 1024 * 128 threads (one per (b, e/4)).
    stack_stream_kernel<<<512, 256, 0, stream>>>(stack, coef, out);
}